// LinearLeakageNlinCore_51505247814101
// MI455X (gfx1250) — compile-verified
//
#include <hip/hip_runtime.h>

typedef float v2f __attribute__((ext_vector_type(2)));
typedef float v8f __attribute__((ext_vector_type(8)));

#define C_CH       64
#define NK         128      // feature dim = 2*C
#define NN         128      // output dim  = 2*C
#define TOK_PER_WG 128
#define LDS_PITCH  130      // pad 128 -> 130 floats: keeps 8B alignment, avoids bank conflicts

// ---------------------------------------------------------------------------
// Kernel 0: expand W[C,2,126] into dense Wfull[n][k] (n = 2c+o), with zeros at
// the "self channel" positions. One thread per element; 16384 elements total.
// ---------------------------------------------------------------------------
__global__ __launch_bounds__(256) void expand_w_kernel(const float* __restrict__ W,
                                                       float* __restrict__ WT) {
    int idx = blockIdx.x * 256 + threadIdx.x;          // 0..16383
    int n = idx >> 7;                                  // output index = 2c+o
    int k = idx & 127;                                 // feature index
    int c = n >> 1;
    int j = k & 63;                                    // source channel
    int half = k >> 6;                                 // 0 = fr block, 1 = fi block
    float v = 0.0f;
    if (j != c) {
        int f = half * 63 + (j < c ? j : j - 1);       // position inside [0,126)
        v = W[n * 126 + f];
    }
    WT[idx] = v;                                       // row-major [n][k]
}

// ---------------------------------------------------------------------------
// Kernel 1: nonlinearity + GEMM via v_wmma_f32_16x16x4_f32.
// Block = 256 threads (8 wave32), handles 128 tokens. Wave w computes the
// 16-token strip [m0, m0+16) x all 128 outputs as 8 accum tiles.
// ---------------------------------------------------------------------------
__global__ __launch_bounds__(256) void leakage_wmma_kernel(const float* __restrict__ x,
                                                           const float* __restrict__ WT,
                                                           float* __restrict__ out) {
    __shared__ float sF[TOK_PER_WG * LDS_PITCH];   // F[token][k], padded pitch
    __shared__ float sW[NN * LDS_PITCH];           // Wfull[n][k], padded pitch

    const int  tid  = threadIdx.x;
    const long tok0 = (long)blockIdx.x * TOK_PER_WG;

    // Stage Wfull into LDS (float2 granularity; 8B-aligned on both sides).
    {
        const float2* g = (const float2*)WT;        // 8192 float2
        for (int i = tid; i < NN * NK / 2; i += 256) {
            int row  = i >> 6;                      // 64 float2 per row
            int col2 = (i & 63) * 2;
            float2 v = g[i];
            sW[row * LDS_PITCH + col2]     = v.x;
            sW[row * LDS_PITCH + col2 + 1] = v.y;
        }
    }

    // Compute F = |z|^2 * z for this block's 128 tokens, write into LDS.
    {
        const float2* x2 = (const float2*)x + tok0 * C_CH;
        for (int i = tid; i < TOK_PER_WG * C_CH; i += 256) {
            int t = i >> 6;
            int c = i & 63;
            float2 v  = x2[i];
            float amp = v.x * v.x + v.y * v.y;
            sF[t * LDS_PITCH + c]        = amp * v.x;   // fr
            sF[t * LDS_PITCH + C_CH + c] = amp * v.y;   // fi
        }
    }
    __syncthreads();

    const int wave  = tid >> 5;
    const int lane  = tid & 31;
    const int lrow  = lane & 15;     // A: row M / B: col N / D: col N
    const int khalf = lane >> 4;     // 0 -> K pair {0,1}; 1 -> K pair {2,3}
    const int m0    = wave * 16;

    v8f acc[8];
    {
        v8f z = {0.f, 0.f, 0.f, 0.f, 0.f, 0.f, 0.f, 0.f};
#pragma unroll
        for (int i = 0; i < 8; ++i) acc[i] = z;
    }

    // Per-lane fragment base pointers (contiguous K pair -> ds_load_b64).
    const float* Abase = &sF[(m0 + lrow) * LDS_PITCH + 2 * khalf];
    const float* Bbase = &sW[lrow * LDS_PITCH + 2 * khalf];

#pragma unroll 4
    for (int kk = 0; kk < NK / 4; ++kk) {
        v2f a = *(const v2f*)(Abase + 4 * kk);
#pragma unroll
        for (int nt = 0; nt < 8; ++nt) {
            v2f b = *(const v2f*)(Bbase + nt * 16 * LDS_PITCH + 4 * kk);
            // (neg_a, A, neg_b, B, c_mod, C, reuse_a, reuse_b)
            acc[nt] = __builtin_amdgcn_wmma_f32_16x16x4_f32(
                false, a, false, b, (short)0, acc[nt], false, false);
        }
    }

    // D tile layout: VGPR r -> row M=r (lanes 0-15) or M=r+8 (lanes 16-31), col N=lrow.
    float* orow = out + (tok0 + m0 + khalf * 8) * NN;
#pragma unroll
    for (int nt = 0; nt < 8; ++nt) {
#pragma unroll
        for (int r = 0; r < 8; ++r) {
            orow[r * NN + nt * 16 + lrow] = acc[nt][r];
        }
    }
}

// ---------------------------------------------------------------------------
// Launch: d_in[0] = x (16*1024*64*2 f32), d_in[1] = W (64*2*126 f32).
// d_out = out (16*1024*64*2 f32). d_ws holds Wfull (64 KB).
// ---------------------------------------------------------------------------
extern "C" void kernel_launch(void* const* d_in, const int* in_sizes, int n_in,
                              void* d_out, int out_size, void* d_ws, size_t ws_size,
                              hipStream_t stream) {
    const float* x  = (const float*)d_in[0];
    const float* W  = (const float*)d_in[1];
    float*       o  = (float*)d_out;
    float*       WT = (float*)d_ws;

    expand_w_kernel<<<64, 256, 0, stream>>>(W, WT);

    const int tokens = 16 * 1024;                       // B*S
    leakage_wmma_kernel<<<tokens / TOK_PER_WG, 256, 0, stream>>>(x, WT, o);
}